// RnnSentiment_8856222565031
// MI455X (gfx1250) — compile-verified
//
#include <hip/hip_runtime.h>
#include <hip/hip_bf16.h>

typedef __attribute__((ext_vector_type(2))) float v2f;
typedef __attribute__((ext_vector_type(8))) float v8f;

#define SEQ    512
#define NBATCH 64
#define HID    256
#define DIM    256
#define FCH    (SEQ * HID)      /* 131072 */
#define NROWS  (SEQ * NBATCH)   /* 32768  */
#define HPAD   (HID + 2)        /* LDS row pad to dodge bank conflicts */

__device__ __forceinline__ v8f wmma_f32x4(v2f a, v2f b, v8f c) {
  // D = A(16x4,f32) * B(4x16,f32) + C(16x16,f32)
  return __builtin_amdgcn_wmma_f32_16x16x4_f32(
      /*neg_a=*/false, a, /*neg_b=*/false, b,
      /*c_mod=*/(short)0, c, /*reuse_a=*/false, /*reuse_b=*/false);
}

// ---------------------------------------------------------------------------
// Phase 1: Xproj[s*64+b, :] = embed[x[s,b]] @ W_ih^T + (b_ih + b_hh)
// One 16x16 output tile per wave; 8 waves per block; 32768 tiles total.
// ---------------------------------------------------------------------------
__global__ __launch_bounds__(256) void xproj_kernel(
    const int* __restrict__ x, const float* __restrict__ embed,
    const float* __restrict__ W_ih, const float* __restrict__ b_ih,
    const float* __restrict__ b_hh, float* __restrict__ xq) {
  const int lane = threadIdx.x & 31;
  const int wave = threadIdx.x >> 5;
  const int l16  = lane & 15;
  const int half = lane >> 4;

  const int tile = blockIdx.x * 8 + wave;   // [0, 32768)
  const int mt   = tile >> 4;               // row tile  [0, 2048)
  const int nt   = tile & 15;               // col tile  [0, 16)

  const int  r = mt * 16 + l16;             // global row = s*64 + b
  const int  n = nt * 16 + l16;             // output column / W_ih row
  const long e = (long)x[r];                // embedding row index

  const float* pa = embed + e * (long)DIM + 2 * half;  // A: row-major gather
  const float* pb = W_ih + (long)n * DIM + 2 * half;   // B = W_ih^T frag

  v8f acc = {};
#pragma unroll 8
  for (int kk = 0; kk < 64; ++kk) {
    v2f a = *(const v2f*)(pa + kk * 4);
    v2f b = *(const v2f*)(pb + kk * 4);
    acc = wmma_f32x4(a, b, acc);
  }

  const float bias = b_ih[n] + b_hh[n];
#pragma unroll
  for (int i = 0; i < 8; ++i) {
    const int m = mt * 16 + i + 8 * half;   // D layout: VGPR i -> M = i + 8*half
    xq[(long)m * HID + n] = acc[i] + bias;
  }
}

// ---------------------------------------------------------------------------
// Phase 2: sequential recurrence, single workgroup (16 waves / 512 threads).
// Wave w owns output columns [16w, 16w+16) for all 64 batch rows.
// W_hh^T fragments live in 128 VGPRs per wave; h lives in LDS.
// Xproj is overwritten in place with ys = tanh(...)  (same [S,B,H] layout).
// ---------------------------------------------------------------------------
__global__ __launch_bounds__(512) void rnn_kernel(
    const float* __restrict__ W_hh, float* __restrict__ xq) {
  __shared__ float hsm[NBATCH][HPAD];

  const int lane = threadIdx.x & 31;
  const int nt   = threadIdx.x >> 5;        // wave id == column tile [0,16)
  const int l16  = lane & 15;
  const int half = lane >> 4;
  const int n    = nt * 16 + l16;

  // B fragments of W_hh^T (constant over all 512 steps) -> registers.
  v2f barr[64];
  {
    const float* pb = W_hh + (long)n * HID + 2 * half;
#pragma unroll
    for (int kk = 0; kk < 64; ++kk) barr[kk] = *(const v2f*)(pb + kk * 4);
  }

  // h0 = 0
  for (int i = threadIdx.x; i < NBATCH * HPAD; i += 512) ((float*)hsm)[i] = 0.0f;
  __syncthreads();

  for (int s = 0; s < SEQ; ++s) {
    v8f acc[4] = {};
    const float* hp = &hsm[l16][2 * half];
#pragma unroll
    for (int kk = 0; kk < 64; ++kk) {
      v2f a0 = *(const v2f*)(hp + 0 * 16 * HPAD + kk * 4);
      v2f a1 = *(const v2f*)(hp + 1 * 16 * HPAD + kk * 4);
      v2f a2 = *(const v2f*)(hp + 2 * 16 * HPAD + kk * 4);
      v2f a3 = *(const v2f*)(hp + 3 * 16 * HPAD + kk * 4);
      acc[0] = wmma_f32x4(a0, barr[kk], acc[0]);
      acc[1] = wmma_f32x4(a1, barr[kk], acc[1]);
      acc[2] = wmma_f32x4(a2, barr[kk], acc[2]);
      acc[3] = wmma_f32x4(a3, barr[kk], acc[3]);
    }
    __syncthreads();   // all waves done reading old h

    float* xrow = xq + (long)s * (NBATCH * HID);
#pragma unroll
    for (int mb = 0; mb < 4; ++mb) {
#pragma unroll
      for (int i = 0; i < 8; ++i) {
        const int m  = mb * 16 + i + 8 * half;   // batch row
        const int ix = m * HID + n;
        const float hv = tanhf(acc[mb][i] + xrow[ix]);
        xrow[ix]   = hv;    // ys stored in place of Xproj
        hsm[m][n]  = hv;    // h for next step
      }
    }
    __syncthreads();   // new h visible before next step reads it
  }
}

// ---------------------------------------------------------------------------
// Phase 3: logits = feat @ fc_W^T + fc_b ; softmax over 2 classes.
// feat[b'] == ys_flat[b'*131072 : (b'+1)*131072] (row-major flatten).
// ---------------------------------------------------------------------------
__global__ __launch_bounds__(256) void fc_softmax_kernel(
    const float* __restrict__ ys, const float* __restrict__ fcW,
    const float* __restrict__ fcb, float* __restrict__ out) {
  __shared__ float r0[256];
  __shared__ float r1[256];
  const int b = blockIdx.x;
  const float* feat = ys + (long)b * FCH;

  float p0 = 0.0f, p1 = 0.0f;
  for (int j = threadIdx.x; j < FCH; j += 256) {
    const float v = feat[j];
    p0 = fmaf(v, fcW[j], p0);
    p1 = fmaf(v, fcW[FCH + j], p1);
  }
  r0[threadIdx.x] = p0;
  r1[threadIdx.x] = p1;
  __syncthreads();
  for (int st = 128; st > 0; st >>= 1) {
    if (threadIdx.x < st) {
      r0[threadIdx.x] += r0[threadIdx.x + st];
      r1[threadIdx.x] += r1[threadIdx.x + st];
    }
    __syncthreads();
  }
  if (threadIdx.x == 0) {
    const float l0 = r0[0] + fcb[0];
    const float l1 = r1[0] + fcb[1];
    const float mx = fmaxf(l0, l1);
    const float e0 = __expf(l0 - mx);
    const float e1 = __expf(l1 - mx);
    const float inv = 1.0f / (e0 + e1);
    out[2 * b]     = e0 * inv;
    out[2 * b + 1] = e1 * inv;
  }
}

// ---------------------------------------------------------------------------
extern "C" void kernel_launch(void* const* d_in, const int* in_sizes, int n_in,
                              void* d_out, int out_size, void* d_ws, size_t ws_size,
                              hipStream_t stream) {
  const int*   x     = (const int*)d_in[0];
  const float* embed = (const float*)d_in[1];
  const float* W_ih  = (const float*)d_in[2];
  const float* W_hh  = (const float*)d_in[3];
  const float* b_ih  = (const float*)d_in[4];
  const float* b_hh  = (const float*)d_in[5];
  const float* fc_W  = (const float*)d_in[6];
  const float* fc_b  = (const float*)d_in[7];
  float*       out   = (float*)d_out;

  float* xq = (float*)d_ws;   // [SEQ, NBATCH, HID] : Xproj, then ys in place (32 MB)

  // Phase 1: 32768 tiles, 8 per block
  xproj_kernel<<<dim3(NROWS * 16 / (16 * 8)), dim3(256), 0, stream>>>(
      x, embed, W_ih, b_ih, b_hh, xq);
  // Phase 2: single-WGP sequential scan
  rnn_kernel<<<dim3(1), dim3(512), 0, stream>>>(W_hh, xq);
  // Phase 3: one block per output row
  fc_softmax_kernel<<<dim3(NBATCH), dim3(256), 0, stream>>>(xq, fc_W, fc_b, out);
}